// GRUModel_78881369358537
// MI455X (gfx1250) — compile-verified
//
#include <hip/hip_runtime.h>

// Problem constants (match reference)
#define B_  4096
#define T_  512
#define F_  8
#define U1_ 8
#define U2_ 16
#define DH_ 16

typedef __attribute__((ext_vector_type(2))) float v2f;
typedef __attribute__((ext_vector_type(8))) float v8f;

static __device__ __forceinline__ v8f wmma4(v2f a, v2f b, v8f c) {
  // D(16x16,f32) = A(16x4,f32) x B(4x16,f32) + C
  return __builtin_amdgcn_wmma_f32_16x16x4_f32(false, a, false, b, (short)0, c,
                                               false, false);
}

// lane-half swap (lane ^ 16) as a single VALU op: v_permlanex16_b32 with
// identity selectors. Avoids ds_bpermute + dscnt waits on the critical path.
static __device__ __forceinline__ float xswap16(float v) {
  int i = __builtin_bit_cast(int, v);
  int r = __builtin_amdgcn_permlanex16(i, i, 0x76543210u, 0xfedcba98u, false, false);
  return __builtin_bit_cast(float, r);
}

static __device__ __forceinline__ float sigmoidf_fast(float x) {
  // 1 / (1 + 2^(-x*log2 e)) -> v_exp_f32 + v_rcp_f32
  return __builtin_amdgcn_rcpf(1.0f + __builtin_amdgcn_exp2f(-1.442695040888963f * x));
}

static __device__ __forceinline__ v8f v8zero() {
  v8f c;
#pragma unroll
  for (int i = 0; i < 8; ++i) c[i] = 0.0f;
  return c;
}

// A-fragment of W^T tile: A[m,k] = W[k,m]; W row-major [nk, ng].
// 16x4 f32 A layout: lanes 0-15 hold K=4kt+0 (v0) / 4kt+1 (v1),
// lanes 16-31 hold K=4kt+2 / 4kt+3; M = lane&15 (+16*mt). Zero-pad OOB.
static __device__ __forceinline__ v2f load_a_frag(const float* W, int nk, int ng,
                                                  int mt, int kt, int lane) {
  int m  = (lane & 15) + 16 * mt;
  int k0 = 4 * kt + ((lane >> 4) << 1);
  v2f a;
  a.x = (m < ng && (k0    ) < nk) ? W[(k0    ) * ng + m] : 0.0f;
  a.y = (m < ng && (k0 + 1) < nk) ? W[(k0 + 1) * ng + m] : 0.0f;
  return a;
}

// Bias as a C fragment: c[v] corresponds to gate (v + 8*laneHalf + 16*mt),
// broadcast across the 16 lanes of each half. Optional second bias added (bi+br).
static __device__ __forceinline__ v8f load_bias_frag(const float* p, const float* q,
                                                     int ng, int mt, int lane) {
  v8f c;
  int base = 8 * (lane >> 4) + 16 * mt;
#pragma unroll
  for (int v = 0; v < 8; ++v) {
    int g = v + base;
    float val = 0.0f;
    if (g < ng) { val = p[g]; if (q) val += q[g]; }
    c[v] = val;
  }
  return c;
}

__global__ __launch_bounds__(128) void gru_rnn_fused(
    const float* __restrict__ x,
    const float* __restrict__ W1, const float* __restrict__ Ur1, const float* __restrict__ b1,
    const float* __restrict__ W2, const float* __restrict__ Ur2, const float* __restrict__ b2,
    const float* __restrict__ Wd, const float* __restrict__ bd,
    const float* __restrict__ Wo, const float* __restrict__ bo,
    float* __restrict__ out)
{
  const int  lane = threadIdx.x & 31;
  const int  wv   = threadIdx.x >> 5;
  const int  tile = blockIdx.x * 4 + wv;   // 256 wave-tiles, 16 batch rows each
  const int  b0   = tile * 16;
  const bool lo   = lane < 16;

  // ---- loop-invariant weight A-fragments (transposed weights) ----
  v2f aW1[2][2], aU1[2][2], aW2[3][2], aU2[3][4], aWd[4];
#pragma unroll
  for (int mt = 0; mt < 2; ++mt)
#pragma unroll
    for (int kt = 0; kt < 2; ++kt) {
      aW1[mt][kt] = load_a_frag(W1,  F_,  3 * U1_, mt, kt, lane);
      aU1[mt][kt] = load_a_frag(Ur1, U1_, 3 * U1_, mt, kt, lane);
    }
#pragma unroll
  for (int mt = 0; mt < 3; ++mt) {
#pragma unroll
    for (int kt = 0; kt < 2; ++kt) aW2[mt][kt] = load_a_frag(W2,  U1_, 3 * U2_, mt, kt, lane);
#pragma unroll
    for (int kt = 0; kt < 4; ++kt) aU2[mt][kt] = load_a_frag(Ur2, U2_, 3 * U2_, mt, kt, lane);
  }
#pragma unroll
  for (int kt = 0; kt < 4; ++kt) aWd[kt] = load_a_frag(Wd, U2_, DH_, 0, kt, lane);

  // ---- biases as C fragments ----
  const v8f cb1_zr = load_bias_frag(b1,           b1 + 3 * U1_, 3 * U1_, 0, lane); // bi+br g0-15
  const v8f cb1_h  = load_bias_frag(b1,           nullptr,      3 * U1_, 1, lane); // bi  g16-23
  const v8f cbr1_h = load_bias_frag(b1 + 3 * U1_, nullptr,      3 * U1_, 1, lane); // br  g16-23
  const v8f cb2_z  = load_bias_frag(b2,           b2 + 3 * U2_, 3 * U2_, 0, lane);
  const v8f cb2_r  = load_bias_frag(b2,           b2 + 3 * U2_, 3 * U2_, 1, lane);
  const v8f cb2_h  = load_bias_frag(b2,           nullptr,      3 * U2_, 2, lane);
  const v8f cbr2_h = load_bias_frag(b2 + 3 * U2_, nullptr,      3 * U2_, 2, lane);
  const v8f cbd    = load_bias_frag(bd,           nullptr,      DH_,     0, lane);
  const float bo0 = bo[0];
  float wo8[8];
#pragma unroll
  for (int v = 0; v < 8; ++v) wo8[v] = Wo[v + 8 * (lane >> 4)];

  // ---- recurrent state in registers ----
  float h1[8];                 // h1 features, valid in lanes 0-15 (batch = lane)
  v8f   h2D = v8zero();        // h2 in D layout: v -> feat v (lo) / v+8 (hi)
  v2f   h1B[2], h2B[4];        // B-operand layouts for next matmuls
#pragma unroll
  for (int v = 0; v < 8; ++v) h1[v] = 0.0f;
#pragma unroll
  for (int i = 0; i < 2; ++i) { h1B[i].x = 0.0f; h1B[i].y = 0.0f; }
#pragma unroll
  for (int i = 0; i < 4; ++i) { h2B[i].x = 0.0f; h2B[i].y = 0.0f; }

  // per-lane x pointer: batch = lane&15, features (lo:0,1 | hi:2,3) then +4
  const float* px = x + (size_t)(b0 + (lane & 15)) * T_ * F_ + (lo ? 0 : 2);

  // software pipeline: x tile for step t is loaded during step t-1
  v2f xb0 = *(const v2f*)(px);
  v2f xb1 = *(const v2f*)(px + 4);

  for (int t = 0; t < T_; ++t) {
    // issue next-step loads now; they retire during this step's WMMA chain
    const float* pn = px + ((t + 1 < T_) ? F_ : 0);
    v2f nb0 = *(const v2f*)(pn);
    v2f nb1 = *(const v2f*)(pn + 4);
    px = pn;

    // ================= GRU1 =================
    // z|r pre-activation: bias(bi+br) + W1^T x + Ur1^T h1, one accumulator
    v8f a1 = wmma4(aW1[0][0], xb0,    cb1_zr);
    a1     = wmma4(aW1[0][1], xb1,    a1);
    a1     = wmma4(aU1[0][0], h1B[0], a1);
    a1     = wmma4(aU1[0][1], h1B[1], a1);
    // hh: gx (bi) and inner (br) kept separate (reset_after)
    v8f g1h = wmma4(aW1[1][0], xb0,    cb1_h);
    g1h     = wmma4(aW1[1][1], xb1,    g1h);
    v8f i1h = wmma4(aU1[1][0], h1B[0], cbr1_h);
    i1h     = wmma4(aU1[1][1], h1B[1], i1h);

#pragma unroll
    for (int v = 0; v < 8; ++v) {
      float zr = sigmoidf_fast(a1[v]);        // lo lanes: z_v, hi lanes: r_v
      float r  = xswap16(zr);                 // bring r to lo lanes
      float hh = fmaxf(g1h[v] + r * i1h[v], 0.0f);
      h1[v]    = zr * h1[v] + (1.0f - zr) * hh;   // valid in lo lanes
    }
    { // rebuild h1 B-fragments (K=0..7): half-swap + select
      float s;
      s = xswap16(h1[2]); h1B[0].x = lo ? h1[0] : s;
      s = xswap16(h1[3]); h1B[0].y = lo ? h1[1] : s;
      s = xswap16(h1[6]); h1B[1].x = lo ? h1[4] : s;
      s = xswap16(h1[7]); h1B[1].y = lo ? h1[5] : s;
    }

    // ================= GRU2 =================
    v8f a2z = wmma4(aW2[0][0], h1B[0], cb2_z);
    a2z     = wmma4(aW2[0][1], h1B[1], a2z);
#pragma unroll
    for (int kt = 0; kt < 4; ++kt) a2z = wmma4(aU2[0][kt], h2B[kt], a2z);
    v8f a2r = wmma4(aW2[1][0], h1B[0], cb2_r);
    a2r     = wmma4(aW2[1][1], h1B[1], a2r);
#pragma unroll
    for (int kt = 0; kt < 4; ++kt) a2r = wmma4(aU2[1][kt], h2B[kt], a2r);
    v8f g2h = wmma4(aW2[2][0], h1B[0], cb2_h);
    g2h     = wmma4(aW2[2][1], h1B[1], g2h);
    v8f i2h = cbr2_h;
#pragma unroll
    for (int kt = 0; kt < 4; ++kt) i2h = wmma4(aU2[2][kt], h2B[kt], i2h);

#pragma unroll
    for (int v = 0; v < 8; ++v) {           // all 16 features, no shuffles needed
      float z  = sigmoidf_fast(a2z[v]);
      float r  = sigmoidf_fast(a2r[v]);
      float hh = fmaxf(g2h[v] + r * i2h[v], 0.0f);
      h2D[v]   = z * h2D[v] + (1.0f - z) * hh;
    }
    { // rebuild h2 B-fragments (K=0..15)
      float s;
      s = xswap16(h2D[2]); h2B[0].x = lo ? h2D[0] : s;
      s = xswap16(h2D[3]); h2B[0].y = lo ? h2D[1] : s;
      s = xswap16(h2D[6]); h2B[1].x = lo ? h2D[4] : s;
      s = xswap16(h2D[7]); h2B[1].y = lo ? h2D[5] : s;
      s = xswap16(h2D[0]); h2B[2].x = lo ? s : h2D[2];
      s = xswap16(h2D[1]); h2B[2].y = lo ? s : h2D[3];
      s = xswap16(h2D[4]); h2B[3].x = lo ? s : h2D[6];
      s = xswap16(h2D[5]); h2B[3].y = lo ? s : h2D[7];
    }

    // ================= dense + output =================
    v8f hd = cbd;
#pragma unroll
    for (int kt = 0; kt < 4; ++kt) hd = wmma4(aWd[kt], h2B[kt], hd);

    float acc = 0.0f;
#pragma unroll
    for (int v = 0; v < 8; ++v) acc += fmaxf(hd[v], 0.0f) * wo8[v];
    acc += xswap16(acc);                    // sum the two lane-halves
    if (lo) out[(size_t)(b0 + lane) * T_ + t] = acc + bo0;

    xb0 = nb0; xb1 = nb1;                   // rotate pipeline
  }
}

extern "C" void kernel_launch(void* const* d_in, const int* in_sizes, int n_in,
                              void* d_out, int out_size, void* d_ws, size_t ws_size,
                              hipStream_t stream) {
  (void)in_sizes; (void)n_in; (void)out_size; (void)d_ws; (void)ws_size;
  const float* x   = (const float*)d_in[0];
  const float* W1  = (const float*)d_in[1];
  const float* Ur1 = (const float*)d_in[2];
  const float* b1  = (const float*)d_in[3];
  const float* W2  = (const float*)d_in[4];
  const float* Ur2 = (const float*)d_in[5];
  const float* b2  = (const float*)d_in[6];
  const float* Wd  = (const float*)d_in[7];
  const float* bd  = (const float*)d_in[8];
  const float* Wo  = (const float*)d_in[9];
  const float* bo  = (const float*)d_in[10];
  float* out = (float*)d_out;

  dim3 grid(B_ / (16 * 4));  // 64 blocks x 4 waves = 256 wave-tiles of 16 rows
  dim3 block(128);
  hipLaunchKernelGGL(gru_rnn_fused, grid, block, 0, stream,
                     x, W1, Ur1, b1, W2, Ur2, b2, Wd, bd, Wo, bo, out);
}